// GridShoot_52956946759702
// MI455X (gfx1250) — compile-verified
//
#include <hip/hip_runtime.h>
#include <math.h>

// ---------------------------------------------------------------------------
// Geodesic shooting (LDDMM) for (2,128,128,128,3) f32 velocity, 8 steps.
// FFT 128 = radix-8 (per lane) x radix-16 (WMMA f32 16x16x4 dense DFT).
// Spectral operator A = d*I + alpha*g*g^H ; Greens via Sherman-Morrison.
// ---------------------------------------------------------------------------

#define NX      128
#define NMASK   127
#define VOL     (NX * NX * NX)       // 2097152
#define NB      2
#define NVOLC   6                    // NB * 3 component volumes
#define NLINES  (NVOLC * NX * NX)    // 98304 1-D lines per pass
#define TWO_PI  6.28318530717958647692f

typedef float v2f __attribute__((ext_vector_type(2)));
typedef float v8f __attribute__((ext_vector_type(8)));
typedef int   v2i __attribute__((ext_vector_type(2)));

#if defined(__AMDGCN__) && __has_builtin(__builtin_amdgcn_global_load_async_to_lds_b64) && \
    __has_builtin(__builtin_amdgcn_global_store_async_from_lds_b64) && \
    __has_builtin(__builtin_amdgcn_s_wait_asynccnt)
#define USE_ASYNC_LDS 1
// b64 async ops take pointers to 64-bit (int2) payloads: global AS(1), LDS AS(3)
#define ASYNC_GPTR(p) ((__attribute__((address_space(1))) v2i*)(p))
#define ASYNC_LPTR(p) ((__attribute__((address_space(3))) v2i*)(p))
#else
#define USE_ASYNC_LDS 0
#endif

#define LPAD 132   // lbuf row stride in float2 (264 dwords == 8 mod 64 -> bank spread)

// ---------------------------------------------------------------------------
// 128-point FFT along AXIS, in place on 6 volumes of 128^3 complex (float2).
// One 128-thread block = 4 waves; 8 lines per block (2 per wave).
// For AXIS 0/1 the 8 lines share x/y and span 8 consecutive z -> cooperative
// z-fastest loads give 64B-contiguous global accesses.
// Stage 1: 16 DFT-8 (per lane) + twiddle.  Stage 2: DFT-16 via WMMA f32.
// ---------------------------------------------------------------------------
template <int AXIS, bool INV>
__global__ __launch_bounds__(128) void fft128_kernel(float2* __restrict__ data) {
    __shared__ float2 lbuf[8][LPAD];     // 8 lines per block
    __shared__ float  yre[4][16][17];    // per-wave stage-1 output [n1][line*8+k2]
    __shared__ float  yim[4][16][17];

    const int tid  = threadIdx.x;
    const int w    = tid >> 5;
    const int lane = tid & 31;
    constexpr float SGN = INV ? 1.0f : -1.0f;

    auto lineBase = [&](int l) -> long {
        int  L   = blockIdx.x * 8 + l;
        int  vol = L >> 14;               // 16384 lines per volume
        int  r   = L & 16383;
        long vb  = (long)vol * VOL;
        if (AXIS == 2)      return vb + (long)r * NX;
        else if (AXIS == 1) { int x = r >> 7, z = r & NMASK; return vb + ((long)x << 14) + z; }
        else                { int y = r >> 7, z = r & NMASK; return vb + ((long)y << 7) + z; }
    };
    const long stride = (AXIS == 2) ? 1 : ((AXIS == 1) ? NX : NX * NX);

    // ---- cooperative load: 8 lines x 128 elements ----
    for (int k = 0; k < 8; ++k) {
        int idx = k * 128 + tid;
        int l   = (AXIS == 2) ? (idx >> 7) : (idx & 7);   // line within block
        int n   = (AXIS == 2) ? (idx & 127) : (idx >> 3); // element within line
        long ga = lineBase(l) + (long)n * stride;
#if USE_ASYNC_LDS
        __builtin_amdgcn_global_load_async_to_lds_b64(
            ASYNC_GPTR(data + ga), ASYNC_LPTR(&lbuf[l][n]), 0, 0);
#else
        lbuf[l][n] = data[ga];
#endif
    }
#if USE_ASYNC_LDS
    __builtin_amdgcn_s_wait_asynccnt(0);
#endif
    __syncthreads();

    // ---- stage 1: DFT-8 over n2 (n = n1 + 16*n2), then twiddle W128^(n1*k2) ----
    {
        const int t  = lane >> 4;       // which of the wave's 2 lines
        const int n1 = lane & 15;
        const int l  = 2 * w + t;
        float xr[8], xi[8];
        for (int m = 0; m < 8; ++m) {
            float2 v = lbuf[l][n1 + 16 * m];
            xr[m] = v.x; xi[m] = v.y;
        }
        // c8[j]=cos(2pi j/8), s8[j]=SGN*sin(2pi j/8): exact constants, SGN folds.
        constexpr float RT2 = 0.70710678118654752440f;
        const float c8[8] = { 1.f,  RT2,       0.f, -RT2,      -1.f, -RT2,       0.f,  RT2 };
        const float s8[8] = { 0.f,  SGN * RT2, SGN, SGN * RT2,  0.f, -SGN * RT2, -SGN, -SGN * RT2 };
        for (int k2 = 0; k2 < 8; ++k2) {
            float ar = 0.f, ai = 0.f;
            for (int m = 0; m < 8; ++m) {
                int q = (m * k2) & 7;
                ar += xr[m] * c8[q] - xi[m] * s8[q];
                ai += xr[m] * s8[q] + xi[m] * c8[q];
            }
            float st, ct;
            __sincosf(SGN * (TWO_PI / 128.0f) * (float)(n1 * k2), &st, &ct); // |q|<=105
            yre[w][n1][t * 8 + k2] = ar * ct - ai * st;
            yim[w][n1][t * 8 + k2] = ar * st + ai * ct;
        }
    }
    __syncthreads();

    // ---- stage 2: C(16x16) = W16(16x16) * Y(16x16), complex via 4 real GEMMs ----
    {
        const int m    = lane & 15;   // A-frag row = output k1
        const int half = lane >> 4;
        const int n    = lane & 15;   // B/C column = (line, k2)
        v8f cr = {0.f, 0.f, 0.f, 0.f, 0.f, 0.f, 0.f, 0.f};
        v8f ci = {0.f, 0.f, 0.f, 0.f, 0.f, 0.f, 0.f, 0.f};
        for (int kc = 0; kc < 4; ++kc) {
            const int kb = kc * 4 + half * 2;          // K indices (kb, kb+1)
            const int p0 = (m * kb) & 15;              // reduce mod 16 before sincos
            const int p1 = (m * (kb + 1)) & 15;
            float s0, c0, s1, c1;
            __sincosf(SGN * (TWO_PI / 16.0f) * (float)p0, &s0, &c0);
            __sincosf(SGN * (TWO_PI / 16.0f) * (float)p1, &s1, &c1);
            v2f awr  = { c0,  c1 };                    // Re(W16)
            v2f awi  = { s0,  s1 };                    // Im(W16) (sign folded in angle)
            v2f awin = { -s0, -s1 };                   // -Im(W16)
            v2f byr  = { yre[w][kb][n], yre[w][kb + 1][n] };
            v2f byi  = { yim[w][kb][n], yim[w][kb + 1][n] };
            cr = __builtin_amdgcn_wmma_f32_16x16x4_f32(false, awr,  false, byr, (short)0, cr, false, false);
            cr = __builtin_amdgcn_wmma_f32_16x16x4_f32(false, awin, false, byi, (short)0, cr, false, false);
            ci = __builtin_amdgcn_wmma_f32_16x16x4_f32(false, awr,  false, byi, (short)0, ci, false, false);
            ci = __builtin_amdgcn_wmma_f32_16x16x4_f32(false, awi,  false, byr, (short)0, ci, false, false);
        }
        __syncthreads();
        const int col = lane & 15;
        const int tt  = col >> 3;
        const int k2  = col & 7;
        for (int v = 0; v < 8; ++v) {
            int k1 = v + 8 * half;                     // C/D layout: M = v + 8*(lane>=16)
            lbuf[2 * w + tt][k2 + 8 * k1] = make_float2(cr[v], ci[v]);
        }
    }
    __syncthreads();

    // ---- cooperative store ----
    for (int k = 0; k < 8; ++k) {
        int idx = k * 128 + tid;
        int l   = (AXIS == 2) ? (idx >> 7) : (idx & 7);
        int n   = (AXIS == 2) ? (idx & 127) : (idx >> 3);
        long ga = lineBase(l) + (long)n * stride;
#if USE_ASYNC_LDS
        __builtin_amdgcn_global_store_async_from_lds_b64(
            ASYNC_GPTR(data + ga), ASYNC_LPTR(&lbuf[l][n]), 0, 0);
#else
        data[ga] = lbuf[l][n];
#endif
    }
#if USE_ASYNC_LDS
    __builtin_amdgcn_s_wait_asynccnt(0);
#endif
}

// ---------------------------------------------------------------------------
// Spectral 3x3 operator per frequency.  A = d*I + alpha*g*g^H (Hermitian
// rank-1 update).  applyA!=0: y = A m ; else y = A^{-1} m (Sherman-Morrison).
// ---------------------------------------------------------------------------
__global__ __launch_bounds__(256) void spectral_kernel(float2* __restrict__ c, int applyA) {
    int idx = blockIdx.x * 256 + threadIdx.x;
    if (idx >= VOL) return;
    int x = idx >> 14, y = (idx >> 7) & NMASK, z = idx & NMASK;

    float sx, cx, sy, cy, sz, cz;
    __sincosf((TWO_PI / 128.0f) * (float)x, &sx, &cx);
    __sincosf((TWO_PI / 128.0f) * (float)y, &sy, &cy);
    __sincosf((TWO_PI / 128.0f) * (float)z, &sz, &cz);

    float L = (2.f - 2.f * cx) + (2.f - 2.f * cy) + (2.f - 2.f * cz);
    float d = 1e-4f + 1e-3f * L + 0.2f * L * L + 0.2f * L;   // ABS+MEM*L+BEND*L^2+SHEAR*L
    const float alpha = 0.25f;                                // SHEAR + DIV
    float2 g0 = { cx - 1.f, sx };
    float2 g1 = { cy - 1.f, sy };
    float2 g2 = { cz - 1.f, sz };
    float n2 = g0.x * g0.x + g0.y * g0.y + g1.x * g1.x + g1.y * g1.y + g2.x * g2.x + g2.y * g2.y;

    float Ac, Bc;       // y = Ac*m + Bc*(s*g), s = g^H m
    if (applyA) { Ac = d;        Bc = alpha; }
    else        { float invd = 1.f / d; Ac = invd; Bc = -alpha / (d + alpha * n2) * invd; }

    for (int b = 0; b < NB; ++b) {
        size_t o0 = (size_t)(b * 3 + 0) * VOL + idx;
        size_t o1 = (size_t)(b * 3 + 1) * VOL + idx;
        size_t o2 = (size_t)(b * 3 + 2) * VOL + idx;
        float2 m0 = c[o0], m1 = c[o1], m2 = c[o2];
        // s = conj(g) . m
        float sr = g0.x * m0.x + g0.y * m0.y + g1.x * m1.x + g1.y * m1.y + g2.x * m2.x + g2.y * m2.y;
        float si = g0.x * m0.y - g0.y * m0.x + g1.x * m1.y - g1.y * m1.x + g2.x * m2.y - g2.y * m2.x;
        sr *= Bc; si *= Bc;
        float2 r0, r1, r2;
        r0.x = Ac * m0.x + (sr * g0.x - si * g0.y); r0.y = Ac * m0.y + (sr * g0.y + si * g0.x);
        r1.x = Ac * m1.x + (sr * g1.x - si * g1.y); r1.y = Ac * m1.y + (sr * g1.y + si * g1.x);
        r2.x = Ac * m2.x + (sr * g2.x - si * g2.y); r2.y = Ac * m2.y + (sr * g2.y + si * g2.x);
        c[o0] = r0; c[o1] = r1; c[o2] = r2;
    }
}

// ---------------------------------------------------------------------------
// Pack (B,X,Y,Z,3) real -> 6 complex volumes ; Extract real part with scale.
// ---------------------------------------------------------------------------
__global__ __launch_bounds__(256) void pack_kernel(const float* __restrict__ src,
                                                   float2* __restrict__ cplx) {
    int i = blockIdx.x * 256 + threadIdx.x;
    if (i >= NB * VOL) return;
    int s = i & (VOL - 1), b = i >> 21;
    for (int c = 0; c < 3; ++c)
        cplx[(size_t)(b * 3 + c) * VOL + s] = make_float2(src[(size_t)i * 3 + c], 0.f);
}

__global__ __launch_bounds__(256) void extract_kernel(const float2* __restrict__ cplx,
                                                      float* __restrict__ dst, float scale) {
    int i = blockIdx.x * 256 + threadIdx.x;
    if (i >= NB * VOL) return;
    int s = i & (VOL - 1), b = i >> 21;
    for (int c = 0; c < 3; ++c)
        dst[(size_t)i * 3 + c] = cplx[(size_t)(b * 3 + c) * VOL + s].x * scale;
}

// ---------------------------------------------------------------------------
// u = disp = velocity / STEPS
// ---------------------------------------------------------------------------
__global__ __launch_bounds__(256) void init_uv_kernel(const float* __restrict__ vel,
                                                      float* __restrict__ u,
                                                      float* __restrict__ disp) {
    int i = blockIdx.x * 256 + threadIdx.x;
    if (i >= NB * VOL * 3) return;
    float v = vel[i] * 0.125f;
    u[i] = v;
    disp[i] = v;
}

// ---------------------------------------------------------------------------
// mt_j = mom_j + sum_i Du[i][j]*mom_i, Du = central diff of u (circular).
// ---------------------------------------------------------------------------
__global__ __launch_bounds__(256) void transport_kernel(const float* __restrict__ u,
                                                        const float* __restrict__ mom,
                                                        float* __restrict__ mt) {
    int i = blockIdx.x * 256 + threadIdx.x;
    if (i >= NB * VOL) return;
    int s = i & (VOL - 1), b = i >> 21;
    int x = s >> 14, y = (s >> 7) & NMASK, z = s & NMASK;
    size_t b3 = (size_t)i * 3;
    size_t bo = (size_t)b * VOL * 3;
    float m0 = mom[b3], m1 = mom[b3 + 1], m2 = mom[b3 + 2];

    size_t sxp = bo + (size_t)((((x + 1) & NMASK) << 14) | (y << 7) | z) * 3;
    size_t sxm = bo + (size_t)((((x - 1) & NMASK) << 14) | (y << 7) | z) * 3;
    size_t syp = bo + (size_t)((x << 14) | (((y + 1) & NMASK) << 7) | z) * 3;
    size_t sym = bo + (size_t)((x << 14) | (((y - 1) & NMASK) << 7) | z) * 3;
    size_t szp = bo + (size_t)((x << 14) | (y << 7) | ((z + 1) & NMASK)) * 3;
    size_t szm = bo + (size_t)((x << 14) | (y << 7) | ((z - 1) & NMASK)) * 3;

    float d00 = 0.5f * (u[sxp + 0] - u[sxm + 0]);
    float d10 = 0.5f * (u[sxp + 1] - u[sxm + 1]);
    float d20 = 0.5f * (u[sxp + 2] - u[sxm + 2]);
    float d01 = 0.5f * (u[syp + 0] - u[sym + 0]);
    float d11 = 0.5f * (u[syp + 1] - u[sym + 1]);
    float d21 = 0.5f * (u[syp + 2] - u[sym + 2]);
    float d02 = 0.5f * (u[szp + 0] - u[szm + 0]);
    float d12 = 0.5f * (u[szp + 1] - u[szm + 1]);
    float d22 = 0.5f * (u[szp + 2] - u[szm + 2]);

    mt[b3 + 0] = m0 + d00 * m0 + d10 * m1 + d20 * m2;
    mt[b3 + 1] = m1 + d01 * m0 + d11 * m1 + d21 * m2;
    mt[b3 + 2] = m2 + d02 * m0 + d12 * m1 + d22 * m2;
}

// ---------------------------------------------------------------------------
// push: scatter-add mt at positions id+u into mom (pre-zeroed), circular.
// ---------------------------------------------------------------------------
__global__ __launch_bounds__(256) void push_kernel(const float* __restrict__ mt,
                                                   const float* __restrict__ u,
                                                   float* __restrict__ mom) {
    int i = blockIdx.x * 256 + threadIdx.x;
    if (i >= NB * VOL) return;
    int s = i & (VOL - 1), b = i >> 21;
    int x = s >> 14, y = (s >> 7) & NMASK, z = s & NMASK;
    size_t b3 = (size_t)i * 3;
    size_t bo = (size_t)b * VOL * 3;

    float v0 = mt[b3], v1 = mt[b3 + 1], v2 = mt[b3 + 2];
    float px = (float)x + u[b3], py = (float)y + u[b3 + 1], pz = (float)z + u[b3 + 2];
    int   gx = (int)floorf(px), gy = (int)floorf(py), gz = (int)floorf(pz);
    float fx = px - (float)gx, fy = py - (float)gy, fz = pz - (float)gz;

    for (int dx = 0; dx < 2; ++dx) {
        float wx = dx ? fx : 1.f - fx;
        int   ix = (gx + dx) & NMASK;
        for (int dy = 0; dy < 2; ++dy) {
            float wy = dy ? fy : 1.f - fy;
            int   iy = (gy + dy) & NMASK;
            for (int dz = 0; dz < 2; ++dz) {
                float wz = dz ? fz : 1.f - fz;
                int   iz = (gz + dz) & NMASK;
                float wgt = wx * wy * wz;
                size_t o = bo + (size_t)((ix << 14) | (iy << 7) | iz) * 3;
                atomicAdd(&mom[o + 0], wgt * v0);
                atomicAdd(&mom[o + 1], wgt * v1);
                atomicAdd(&mom[o + 2], wgt * v2);
            }
        }
    }
}

// ---------------------------------------------------------------------------
// pull+add: disp[v] += trilinear(u, v + disp[v]), circular.
// ---------------------------------------------------------------------------
__global__ __launch_bounds__(256) void pull_add_kernel(const float* __restrict__ u,
                                                       float* __restrict__ disp) {
    int i = blockIdx.x * 256 + threadIdx.x;
    if (i >= NB * VOL) return;
    int s = i & (VOL - 1), b = i >> 21;
    int x = s >> 14, y = (s >> 7) & NMASK, z = s & NMASK;
    size_t b3 = (size_t)i * 3;
    size_t bo = (size_t)b * VOL * 3;

    float px = (float)x + disp[b3], py = (float)y + disp[b3 + 1], pz = (float)z + disp[b3 + 2];
    int   gx = (int)floorf(px), gy = (int)floorf(py), gz = (int)floorf(pz);
    float fx = px - (float)gx, fy = py - (float)gy, fz = pz - (float)gz;

    float a0 = 0.f, a1 = 0.f, a2 = 0.f;
    for (int dx = 0; dx < 2; ++dx) {
        float wx = dx ? fx : 1.f - fx;
        int   ix = (gx + dx) & NMASK;
        for (int dy = 0; dy < 2; ++dy) {
            float wy = dy ? fy : 1.f - fy;
            int   iy = (gy + dy) & NMASK;
            for (int dz = 0; dz < 2; ++dz) {
                float wz = dz ? fz : 1.f - fz;
                int   iz = (gz + dz) & NMASK;
                float wgt = wx * wy * wz;
                size_t o = bo + (size_t)((ix << 14) | (iy << 7) | iz) * 3;
                a0 += wgt * u[o + 0];
                a1 += wgt * u[o + 1];
                a2 += wgt * u[o + 2];
            }
        }
    }
    disp[b3 + 0] += a0;
    disp[b3 + 1] += a1;
    disp[b3 + 2] += a2;
}

// ---------------------------------------------------------------------------
// out = id + disp
// ---------------------------------------------------------------------------
__global__ __launch_bounds__(256) void final_kernel(const float* __restrict__ disp,
                                                    float* __restrict__ out) {
    int i = blockIdx.x * 256 + threadIdx.x;
    if (i >= NB * VOL) return;
    int s = i & (VOL - 1);
    int x = s >> 14, y = (s >> 7) & NMASK, z = s & NMASK;
    size_t b3 = (size_t)i * 3;
    out[b3 + 0] = (float)x + disp[b3];
    out[b3 + 1] = (float)y + disp[b3 + 1];
    out[b3 + 2] = (float)z + disp[b3 + 2];
}

// ---------------------------------------------------------------------------
// Host orchestration (all on `stream`; graph-capture safe).
// ---------------------------------------------------------------------------
extern "C" void kernel_launch(void* const* d_in, const int* in_sizes, int n_in,
                              void* d_out, int out_size, void* d_ws, size_t ws_size,
                              hipStream_t stream) {
    (void)in_sizes; (void)n_in; (void)out_size; (void)ws_size;
    const float* vel = (const float*)d_in[0];
    float*       out = (float*)d_out;

    char*   ws    = (char*)d_ws;
    float2* cplx  = (float2*)ws;                          // 6 * VOL * 8B  = ~100.7 MB
    size_t  cby   = (size_t)NVOLC * VOL * sizeof(float2);
    size_t  fby   = (size_t)NB * VOL * 3 * sizeof(float); // ~50.3 MB each
    float*  mom   = (float*)(ws + cby);
    float*  u     = (float*)(ws + cby + fby);
    float*  disp  = (float*)(ws + cby + 2 * fby);
    float*  mt    = (float*)(ws + cby + 3 * fby);

    const dim3 fftGrid(NLINES / 8), fftBlk(128);
    const int  nPt = NB * VOL;                 // 4,194,304
    const dim3 ptG((nPt + 255) / 256), ptB(256);
    const dim3 spG((VOL + 255) / 256);
    const dim3 inG((NB * VOL * 3 + 255) / 256);

    auto fwd_fft = [&]() {
        fft128_kernel<2, false><<<fftGrid, fftBlk, 0, stream>>>(cplx);
        fft128_kernel<1, false><<<fftGrid, fftBlk, 0, stream>>>(cplx);
        fft128_kernel<0, false><<<fftGrid, fftBlk, 0, stream>>>(cplx);
    };
    auto inv_fft = [&]() {
        fft128_kernel<2, true><<<fftGrid, fftBlk, 0, stream>>>(cplx);
        fft128_kernel<1, true><<<fftGrid, fftBlk, 0, stream>>>(cplx);
        fft128_kernel<0, true><<<fftGrid, fftBlk, 0, stream>>>(cplx);
    };

    // mom = apply_op(velocity, A)
    pack_kernel<<<ptG, ptB, 0, stream>>>(vel, cplx);
    fwd_fft();
    spectral_kernel<<<spG, ptB, 0, stream>>>(cplx, 1);
    inv_fft();
    extract_kernel<<<ptG, ptB, 0, stream>>>(cplx, mom, 1.0f / (float)VOL);

    // u = disp = velocity / STEPS
    init_uv_kernel<<<inG, ptB, 0, stream>>>(vel, u, disp);

    for (int step = 1; step < 8; ++step) {
        // mt = (I + Du)^T mom
        transport_kernel<<<ptG, ptB, 0, stream>>>(u, mom, mt);
        // mom = push(mt, id + u)
        (void)hipMemsetAsync(mom, 0, fby, stream);
        push_kernel<<<ptG, ptB, 0, stream>>>(mt, u, mom);
        // u = apply_op(mom, Greens) / STEPS
        pack_kernel<<<ptG, ptB, 0, stream>>>(mom, cplx);
        fwd_fft();
        spectral_kernel<<<spG, ptB, 0, stream>>>(cplx, 0);
        inv_fft();
        extract_kernel<<<ptG, ptB, 0, stream>>>(cplx, u, 1.0f / ((float)VOL * 8.0f));
        // disp += pull(u, id + disp)
        pull_add_kernel<<<ptG, ptB, 0, stream>>>(u, disp);
    }

    final_kernel<<<ptG, ptB, 0, stream>>>(disp, out);
}